// BatchedDynamicEmbeddingTablesV2_45432164057624
// MI455X (gfx1250) — compile-verified
//
#include <hip/hip_runtime.h>
#include <hip/hip_bf16.h>

typedef __attribute__((ext_vector_type(2))) float v2f;
typedef __attribute__((ext_vector_type(8))) float v8f;

#define CAPACITY        2097152
#define DIM             64
#define BATCH           16384
#define BAG             50
#define BAGS_PER_BLOCK  16
#define ROWS_PER_BLOCK  (BAGS_PER_BLOCK * BAG)   /* 800 rows per 16-bag group */
#define ROWS_PER_SEG    (ROWS_PER_BLOCK / 2)     /* 400: split across 2 wave groups */
#define KSTEPS          (ROWS_PER_SEG / 4)       /* 100 WMMAs per wave */

// One block: 16 bags x 64 cols, 8 waves. Waves 0-3 own column tiles for rows
// 0-399 of the group; waves 4-7 the same tiles for rows 400-799. Each
// V_WMMA_F32_16X16X4_F32 consumes 4 gathered rows (k-slots); A is a 0/1
// selection matrix routing row-slot k to its bag. Partials combine via LDS.
__global__ __launch_bounds__(256, 4)
void emb_bag_pool_wmma(const float* __restrict__ values,
                       const int*   __restrict__ indices,
                       float*       __restrict__ out) {
    __shared__ unsigned s_off[ROWS_PER_BLOCK];     // pre-shifted row byte offsets
    __shared__ float    s_part[4][16][16];         // [coltile][M][N] partials

    const int tid  = threadIdx.x;
    const int wave = tid >> 5;
    const int cw   = wave & 3;    // column tile 0..3
    const int seg  = wave >> 2;   // row segment 0..1
    const int lane = tid & 31;
    const int half = lane >> 4;   // 0: K0/K1 slots, 1: K2/K3 slots
    const int m    = lane & 15;   // M (bag) for A-lanes, N (col) for B/D-lanes
    const int blk  = blockIdx.x;

    // Stage this block's 800 row byte-offsets into LDS (idx*256 fits u32).
    const long long ibase = (long long)blk * ROWS_PER_BLOCK;
    for (int i = tid; i < ROWS_PER_BLOCK; i += 256)
        s_off[i] = ((unsigned)indices[ibase + i]) << 8;
    __syncthreads();

    // Wave-uniform scalar base (column tile) -> SADDR + u32-voffset gathers.
    const int colbase = __builtin_amdgcn_readfirstlane(cw * 16);
    const char* __restrict__ base = (const char*)(values + colbase);
    const unsigned moff = (unsigned)(m << 2);
    const int koff = seg * ROWS_PER_SEG;           // this wave's k-slot origin

    v8f acc = {};

    // Incremental selection state: d = kk0 - 50*m for this lane's first
    // k-slot (kk0 = koff + 4t + 2*half). A[m,kk] = 1 <=> 0 <= kk-50m < 50.
    int d = koff + 2 * half - BAG * m;

    auto load_trip = [&](int t, float* bb) {
#pragma unroll
        for (int u = 0; u < 4; ++u) {
            const int k = koff + 4 * (t + u) + 2 * half;
            const unsigned o0 = s_off[k]     + moff;
            const unsigned o1 = s_off[k + 1] + moff;
            bb[2 * u]     = *(const float*)(base + o0);
            bb[2 * u + 1] = *(const float*)(base + o1);
        }
    };

    auto mm_trip = [&](const float* bb) {
#pragma unroll
        for (int u = 0; u < 4; ++u) {
            const int du = d + 4 * u;
            v2f a;
            a.x = ((unsigned)du       < (unsigned)BAG) ? 1.0f : 0.0f;
            a.y = ((unsigned)(du + 1) < (unsigned)BAG) ? 1.0f : 0.0f;
            v2f b;
            b.x = bb[2 * u];
            b.y = bb[2 * u + 1];
            acc = __builtin_amdgcn_wmma_f32_16x16x4_f32(
                /*neg_a=*/false, a, /*neg_b=*/false, b,
                /*c_mod=*/(short)0, acc, /*reuse_a=*/false, /*reuse_b=*/false);
        }
        d += 16;
    };

    // Depth-3 software pipeline: 16 gathers in flight while WMMAs consume.
    float b0[8], b1[8], b2[8];
    load_trip(0, b0);
    load_trip(4, b1);
    for (int t = 0; t + 8 < KSTEPS; t += 4) {
        load_trip(t + 8, b2);
        mm_trip(b0);
#pragma unroll
        for (int i = 0; i < 8; ++i) { b0[i] = b1[i]; b1[i] = b2[i]; }
    }
    mm_trip(b0);
    mm_trip(b1);

    // Combine the two row-segments: seg 1 parks its D tile in LDS, seg 0
    // adds it (fixed low+high order -> deterministic) and stores.
    if (seg == 1) {
#pragma unroll
        for (int p = 0; p < 8; ++p)
            s_part[cw][p + 8 * half][m] = acc[p];
    }
    __syncthreads();
    if (seg == 0) {
#pragma unroll
        for (int p = 0; p < 8; ++p)
            acc[p] += s_part[cw][p + 8 * half][m];

        // D layout: lane holds D[M = p + 8*half, N = m] in acc[p].
        const long long bag0 = (long long)blk * BAGS_PER_BLOCK + 8 * half;
        float* __restrict__ o = out + bag0 * DIM + colbase + m;
#pragma unroll
        for (int p = 0; p < 8; ++p)
            o[(long long)p * DIM] = acc[p];
    }
}

extern "C" void kernel_launch(void* const* d_in, const int* in_sizes, int n_in,
                              void* d_out, int out_size, void* d_ws, size_t ws_size,
                              hipStream_t stream) {
    const float* values  = (const float*)d_in[0];
    const int*   indices = (const int*)d_in[1];
    float*       out     = (float*)d_out;

    dim3 grid(BATCH / BAGS_PER_BLOCK);  // 1024
    dim3 block(256);                    // 8 waves: 4 column tiles x 2 row segs
    emb_bag_pool_wmma<<<grid, block, 0, stream>>>(values, indices, out);
}